// CostVolume_78451872629268
// MI455X (gfx1250) — compile-verified
//
#include <hip/hip_runtime.h>

typedef __attribute__((ext_vector_type(2))) float v2f;
typedef __attribute__((ext_vector_type(8))) float v8f;

#define BATCH 4
#define CH    320
#define HH    128
#define WW    240
#define DD    49                    // maxdisp + 1
#define HW    (HH * WW)             // 30720
#define CHW   ((size_t)CH * HW)     // per-batch stride
#define BHW   (BATCH * HH * WW)     // stride between disparity planes
#define NTW   (WW / 16)             // 15 row tiles along w
#define NJOBS (BATCH * HH * NTW)    // 7680 wave-jobs

// Zero-fill output (covers the w < d padding region; valid region overwritten).
__global__ __launch_bounds__(256) void cv_zero(float* __restrict__ out, int n) {
    int idx = blockIdx.x * 256 + threadIdx.x;
    if (idx < n) out[idx] = 0.0f;
}

// One wave per BLOCK (blockDim=32): all job decomposition is scalar (SGPR).
// Each wave computes the <=4 banded 16x16 tiles (column tiles j = i-3..i) of
// the per-(b,h) banded GEMM X * Y^T, K = C = 320, via V_WMMA_F32_16X16X4_F32.
__global__ __launch_bounds__(32) void cv_wmma(const float* __restrict__ x,
                                              const float* __restrict__ y,
                                              float* __restrict__ out) {
    // ---- scalar (wave-uniform) job decomposition ----
    const int job = blockIdx.x;             // 0 .. NJOBS-1, scalar
    const int i   = job % NTW;              // row tile (w block)
    const int bh  = job / NTW;
    const int h   = bh % HH;
    const int b   = bh / HH;

    const size_t bhoff = (size_t)b * CHW + (size_t)h * WW;
    const int wbase  = i * 16;
    const int jstart = i - 3;
    const int outbase = (b * HH + h) * WW;  // scalar part of store offset

    // ---- per-lane fragment coordinates ----
    const int lane = threadIdx.x & 31;
    const int half = lane >> 4;             // 0: lanes 0-15, 1: lanes 16-31
    const int l    = lane & 15;

    // A fragment source: X[w = wbase+l][c], c striped as K = 2*half + vgpr.
    const float* px = x + bhoff + (size_t)(2 * half) * HW + wbase + l;

    // B fragment sources per diagonal tile: Y[w' = 16*j + l][c], same striping.
    const float* py[4];
    bool valid[4];
#pragma unroll
    for (int jo = 0; jo < 4; ++jo) {
        const int j = jstart + jo;
        valid[jo] = (j >= 0);               // scalar
        const int wpb = valid[jo] ? j * 16 : 0;
        py[jo] = y + bhoff + (size_t)(2 * half) * HW + wpb + l;
    }

    v8f acc[4] = {};

#pragma unroll 8
    for (int k0 = 0; k0 < CH; k0 += 4) {
        v2f a;
        a.x = px[0];        // c = k0 + 2*half
        a.y = px[HW];       // c = k0 + 2*half + 1
        px += (size_t)4 * HW;

#pragma unroll
        for (int jo = 0; jo < 4; ++jo) {
            if (valid[jo]) {                // scalar branch
                v2f bb;
                bb.x = py[jo][0];
                bb.y = py[jo][HW];
                acc[jo] = __builtin_amdgcn_wmma_f32_16x16x4_f32(
                    /*neg_a=*/false, a, /*neg_b=*/false, bb,
                    /*c_mod=*/(short)0, acc[jo],
                    /*reuse_a=*/false, /*reuse_b=*/false);
            }
            py[jo] += (size_t)4 * HW;
        }
    }

    // D layout: VGPR r -> M = r + 8*half, N = l. Store only band 0 <= d <= 48.
#pragma unroll
    for (int jo = 0; jo < 4; ++jo) {
        if (!valid[jo]) continue;           // scalar branch
        const int wp = (jstart + jo) * 16 + l;      // w' = w - d
#pragma unroll
        for (int r = 0; r < 8; ++r) {
            const int w = wbase + r + 8 * half;
            const int d = w - wp;
            if (d >= 0 && d < DD) {
                out[(size_t)d * BHW + outbase + w] = acc[jo][r];
            }
        }
    }
}

extern "C" void kernel_launch(void* const* d_in, const int* in_sizes, int n_in,
                              void* d_out, int out_size, void* d_ws, size_t ws_size,
                              hipStream_t stream) {
    const float* x = (const float*)d_in[0];
    const float* y = (const float*)d_in[1];
    float* out = (float*)d_out;

    const int n = out_size;  // 49*4*128*240 = 6,021,120
    cv_zero<<<(n + 255) / 256, 256, 0, stream>>>(out, n);

    // One wave per block: job decomposition is fully scalar.
    cv_wmma<<<NJOBS, 32, 0, stream>>>(x, y, out);
}